// MSDeformableAttention_3813930958893
// MI455X (gfx1250) — compile-verified
//
#include <hip/hip_runtime.h>
#include <hip/hip_bf16.h>

typedef __attribute__((ext_vector_type(2))) float v2f;
typedef __attribute__((ext_vector_type(8))) float v8f;

#define BSZ 2
#define NQ  10000
#define CCH 256
#define MH  8
#define LL  3
#define KP  4
#define DD  32

// ---------------------------------------------------------------------------
// N-register-blocked GEMM with compile-time K (fully unrolled):
//   Cout[m][n] = sum_k A[m][k] * W[n][k] + bias[n]     (C = A @ W^T + b)
// A row-major (Mdim x KD), W row-major (Ndim x KD).
// One wave32 computes a 16 x (16*NB) tile via V_WMMA_F32_16X16X4_F32.
// Per K-step: 1 A-fragment load shared across NB WMMAs. Full unroll lets the
// backend use fixed base pointers + immediate offsets and s_clause'd loads.
// grid.x = Mdim/16, grid.y = Ndim/(16*NB) (both must divide), block = 32.
// ---------------------------------------------------------------------------
template <int NB, int KD>
__global__ void msda_gemm_qwt(const float* __restrict__ A,
                              const float* __restrict__ W,
                              const float* __restrict__ bias,
                              float* __restrict__ Cout,
                              int Ndim) {
  const int tm   = blockIdx.x * 16;
  const int tn   = blockIdx.y * (16 * NB);
  const int lane = threadIdx.x & 31;
  const int rc   = lane & 15;          // A-row / B-col index for this lane
  const int kb   = (lane >> 4) * 2;    // lanes 16-31 hold K+2,K+3

  const float* Ap = A + (size_t)((tm + rc) * KD + kb);
  const float* Wp[NB];
#pragma unroll
  for (int j = 0; j < NB; ++j) Wp[j] = W + (size_t)((tn + j * 16 + rc) * KD + kb);

  v8f c[NB] = {};
#pragma unroll
  for (int k = 0; k < KD; k += 4) {
    const v2f a = *(const v2f*)(Ap + k);
#pragma unroll
    for (int j = 0; j < NB; ++j) {
      const v2f b = *(const v2f*)(Wp[j] + k);
      c[j] = __builtin_amdgcn_wmma_f32_16x16x4_f32(false, a, false, b,
                                                   (short)0, c[j], false, false);
    }
  }

  const int mofs = (lane >> 4) * 8;    // lanes 16-31 hold tile rows 8..15
#pragma unroll
  for (int j = 0; j < NB; ++j) {
    const int colN = tn + j * 16 + rc;
    const float bn = bias[colN];
#pragma unroll
    for (int r = 0; r < 8; ++r) {
      Cout[(size_t)(tm + mofs + r) * Ndim + colN] = c[j][r] + bn;
    }
  }
}

// ---------------------------------------------------------------------------
// Value projection (1x1 conv) for one level (HW compile-time):
//   proj[b][m][hw][d] = sum_c Wv[m*32+d][c] * V[b][c][hw] + bv[m*32+d]
// GEMM rows = hw (lane-coalesced, K-stride = HW), cols = out-channel,
// contraction over C=256. NB-blocked over output channels; fully unrolled.
// grid.x = ceil(HW/16), grid.y = 256/(16*NB), grid.z = B, block = 32.
// ---------------------------------------------------------------------------
template <int NB, int HW>
__global__ void msda_value_proj(const float* __restrict__ V,
                                const float* __restrict__ Wv,
                                const float* __restrict__ bv,
                                float* __restrict__ proj) {
  const int thw  = blockIdx.x * 16;
  const int to   = blockIdx.y * (16 * NB);
  const int bb   = blockIdx.z;
  const int lane = threadIdx.x & 31;
  const int rc   = lane & 15;
  const int kb   = (lane >> 4) * 2;

  const int hwA  = min(thw + rc, HW - 1);        // clamp ragged tail reads
  const float* VpA = V + (size_t)bb * CCH * HW + (size_t)(kb * HW + hwA);
  const float* Wp[NB];
#pragma unroll
  for (int j = 0; j < NB; ++j) Wp[j] = Wv + (size_t)((to + j * 16 + rc) * CCH + kb);

  v8f c[NB] = {};
#pragma unroll
  for (int k = 0; k < CCH; k += 4) {
    v2f a;
    a.x = VpA[(size_t)k * HW];
    a.y = VpA[(size_t)k * HW + HW];
#pragma unroll
    for (int j = 0; j < NB; ++j) {
      const v2f b = *(const v2f*)(Wp[j] + k);
      c[j] = __builtin_amdgcn_wmma_f32_16x16x4_f32(false, a, false, b,
                                                   (short)0, c[j], false, false);
    }
  }

  const int mofs = (lane >> 4) * 8;
#pragma unroll
  for (int j = 0; j < NB; ++j) {
    const int o = to + j * 16 + rc;     // output channel held by this lane
    const int m = o >> 5;
    const int d = o & 31;
    const float bo = bv[o];
    float* pb = proj + ((size_t)bb * MH + m) * (size_t)HW * DD + d;
#pragma unroll
    for (int r = 0; r < 8; ++r) {
      int hw = thw + mofs + r;
      if (hw < HW) pb[(size_t)hw * DD] = c[j][r] + bo;
    }
  }
}

// ---------------------------------------------------------------------------
// Softmax over the 12 (L*K) logits per (b, q, m). In-place, one thread each.
// ---------------------------------------------------------------------------
__global__ void msda_softmax12(float* __restrict__ attn) {
  const int t = blockIdx.x * blockDim.x + threadIdx.x;
  if (t >= BSZ * NQ * MH) return;
  float* p = attn + (size_t)(t >> 3) * (MH * LL * KP) + (t & 7) * (LL * KP);
  float mx = p[0];
#pragma unroll
  for (int j = 1; j < 12; ++j) mx = fmaxf(mx, p[j]);
  float e[12];
  float s = 0.f;
#pragma unroll
  for (int j = 0; j < 12; ++j) { e[j] = expf(p[j] - mx); s += e[j]; }
  const float inv = 1.f / s;
#pragma unroll
  for (int j = 0; j < 12; ++j) p[j] = e[j] * inv;
}

// ---------------------------------------------------------------------------
// Bilinear sampling + attention-weighted accumulation.
// One wave32 per (b, q, m); lane = channel d. Each corner gather is a
// coalesced 128B read (proj layout [b][m][hw][d]).
// ---------------------------------------------------------------------------
__global__ void msda_sample(const float* __restrict__ refp,
                            const float* __restrict__ off,
                            const float* __restrict__ attn,
                            const float* __restrict__ p0,
                            const float* __restrict__ p1,
                            const float* __restrict__ p2,
                            float* __restrict__ mid) {
  const int wid = (int)((blockIdx.x * blockDim.x + threadIdx.x) >> 5);
  const int d   = threadIdx.x & 31;
  const int m   = wid & 7;
  const int bq  = wid >> 3;            // b*NQ + q
  const int b   = bq / NQ;

  const float* ob = off  + (size_t)bq * (MH * LL * KP * 2) + m * (LL * KP * 2);
  const float* ab = attn + (size_t)bq * (MH * LL * KP)     + m * (LL * KP);
  const float rx = refp[(size_t)bq * 2 + 0];
  const float ry = refp[(size_t)bq * 2 + 1];

  const float* projs[3] = {p0, p1, p2};
  const int    Hs[3]    = {100, 50, 25};

  float acc = 0.f;
#pragma unroll
  for (int l = 0; l < LL; ++l) {
    const int Hh = Hs[l];
    const int Ww = Hh;
    const int HW = Hh * Ww;
    const float* pl = projs[l] + ((size_t)b * MH + m) * (size_t)HW * DD + d;
#pragma unroll
    for (int kk = 0; kk < KP; ++kk) {
      const float ox = ob[(l * KP + kk) * 2 + 0];
      const float oy = ob[(l * KP + kk) * 2 + 1];
      const float aw = ab[l * KP + kk];
      // loc = ref + off/size; pixel = loc*size - 0.5
      const float px = (rx + ox / (float)Ww) * (float)Ww - 0.5f;
      const float py = (ry + oy / (float)Hh) * (float)Hh - 0.5f;
      const float x0f = floorf(px), y0f = floorf(py);
      const float wx1 = px - x0f,   wy1 = py - y0f;
      const float wx0 = 1.f - wx1,  wy0 = 1.f - wy1;
      const int x0 = (int)x0f, y0 = (int)y0f;
      const int x1 = x0 + 1,   y1 = y0 + 1;
      float v00 = 0.f, v10 = 0.f, v01 = 0.f, v11 = 0.f;
      const bool xi0 = (x0 >= 0) & (x0 < Ww);
      const bool xi1 = (x1 >= 0) & (x1 < Ww);
      const bool yi0 = (y0 >= 0) & (y0 < Hh);
      const bool yi1 = (y1 >= 0) & (y1 < Hh);
      if (xi0 & yi0) v00 = pl[(size_t)(y0 * Ww + x0) * DD];
      if (xi1 & yi0) v10 = pl[(size_t)(y0 * Ww + x1) * DD];
      if (xi0 & yi1) v01 = pl[(size_t)(y1 * Ww + x0) * DD];
      if (xi1 & yi1) v11 = pl[(size_t)(y1 * Ww + x1) * DD];
      acc += aw * (v00 * wx0 * wy0 + v10 * wx1 * wy0 +
                   v01 * wx0 * wy1 + v11 * wx1 * wy1);
    }
  }
  mid[(size_t)bq * CCH + m * DD + d] = acc;
}

// ---------------------------------------------------------------------------
extern "C" void kernel_launch(void* const* d_in, const int* in_sizes, int n_in,
                              void* d_out, int out_size, void* d_ws, size_t ws_size,
                              hipStream_t stream) {
  const float* query = (const float*)d_in[0];   // (B, NQ, 256)
  const float* refp  = (const float*)d_in[1];   // (B, NQ, 2)
  const float* val0  = (const float*)d_in[2];   // (B, 256, 100, 100)
  const float* val1  = (const float*)d_in[3];   // (B, 256, 50, 50)
  const float* val2  = (const float*)d_in[4];   // (B, 256, 25, 25)
  const float* Wv    = (const float*)d_in[5];   // (256, 256)
  const float* bv    = (const float*)d_in[6];   // (256,)
  const float* Ws    = (const float*)d_in[7];   // (192, 256)
  const float* bs    = (const float*)d_in[8];   // (192,)
  const float* Wa    = (const float*)d_in[9];   // (96, 256)
  const float* ba    = (const float*)d_in[10];  // (96,)
  const float* Wo    = (const float*)d_in[11];  // (256, 256)
  const float* bo    = (const float*)d_in[12];  // (256,)
  float* out = (float*)d_out;                   // (B, NQ, 256)

  // Workspace carve-up (floats)
  float* ws = (float*)d_ws;
  const size_t nRows = (size_t)BSZ * NQ;              // 20000
  float* ws_off  = ws;                                // 20000*192
  float* ws_attn = ws_off  + nRows * 192;             // 20000*96
  float* ws_p0   = ws_attn + nRows * 96;              // 2*8*10000*32
  float* ws_p1   = ws_p0   + (size_t)BSZ * MH * 10000 * DD;
  float* ws_p2   = ws_p1   + (size_t)BSZ * MH * 2500  * DD;
  float* ws_mid  = ws_p2   + (size_t)BSZ * MH * 625   * DD;  // 20000*256

  // 1) sampling offsets: off = query @ Ws^T + bs   (20000 x 192), NB=4 -> 3 n-blocks
  msda_gemm_qwt<4, 256><<<dim3(1250, 192 / 64), 32, 0, stream>>>(query, Ws, bs, ws_off, 192);

  // 2) attention logits: query @ Wa^T + ba         (20000 x 96),  NB=2 -> 3 n-blocks
  msda_gemm_qwt<2, 256><<<dim3(1250, 96 / 32), 32, 0, stream>>>(query, Wa, ba, ws_attn, 96);

  // 3) softmax over L*K=12 per (b,q,m)
  msda_softmax12<<<(BSZ * NQ * MH + 255) / 256, 256, 0, stream>>>(ws_attn);

  // 4) per-level value projection -> [b][m][hw][d], NB=4 -> 4 channel-blocks
  msda_value_proj<4, 10000><<<dim3((10000 + 15) / 16, 4, BSZ), 32, 0, stream>>>(val0, Wv, bv, ws_p0);
  msda_value_proj<4, 2500 ><<<dim3((2500  + 15) / 16, 4, BSZ), 32, 0, stream>>>(val1, Wv, bv, ws_p1);
  msda_value_proj<4, 625  ><<<dim3((625   + 15) / 16, 4, BSZ), 32, 0, stream>>>(val2, Wv, bv, ws_p2);

  // 5) bilinear sampling + attention accumulate -> mid (20000 x 256)
  msda_sample<<<(BSZ * NQ * MH) / 8, 256, 0, stream>>>(refp, ws_off, ws_attn,
                                                       ws_p0, ws_p1, ws_p2, ws_mid);

  // 6) output projection: out = mid @ Wo^T + bo    (20000 x 256), NB=4 -> 4 n-blocks
  msda_gemm_qwt<4, 256><<<dim3(1250, 256 / 64), 32, 0, stream>>>(ws_mid, Wo, bo, out, 256);
}